// QueryGenerator_15410342658359
// MI455X (gfx1250) — compile-verified
//
#include <hip/hip_runtime.h>
#include <hip/hip_bf16.h>

typedef __attribute__((ext_vector_type(2))) float v2f;
typedef __attribute__((ext_vector_type(8))) float v8f;

#define B_   256
#define S_   8
#define Q_   4
#define H_   256
#define N_   32
#define G_   4096
#define H2_  512
#define KB   32
#define LDA  36   // padded LDS row stride (floats); 36*d mod 64 banks -> conflict-free b64 frag reads

// ---------------------------------------------------------------------------
// Phase 1: fused concat + LayerNorm.  xn[b,g] = (gi - mu) * rsqrt(var + eps)
// ---------------------------------------------------------------------------
__global__ __launch_bounds__(256) void ln_kernel(const float* __restrict__ ns,
                                                 const float* __restrict__ sp,
                                                 float* __restrict__ xn) {
  const int b = blockIdx.x;
  const int t = threadIdx.x;
  const float* src = (t < 128) ? (ns + (size_t)b * 2048 + t * 16)
                               : (sp + (size_t)b * 2048 + (t - 128) * 16);
  float4 v[4];
  float s = 0.f, s2 = 0.f;
#pragma unroll
  for (int j = 0; j < 4; ++j) {
    v[j] = *(const float4*)(src + j * 4);
    s  += v[j].x + v[j].y + v[j].z + v[j].w;
    s2 += v[j].x * v[j].x + v[j].y * v[j].y + v[j].z * v[j].z + v[j].w * v[j].w;
  }
#pragma unroll
  for (int m = 16; m > 0; m >>= 1) {
    s  += __shfl_xor(s,  m, 32);
    s2 += __shfl_xor(s2, m, 32);
  }
  __shared__ float red[2][8];
  if ((t & 31) == 0) { red[0][t >> 5] = s; red[1][t >> 5] = s2; }
  __syncthreads();
  float ts = 0.f, ts2 = 0.f;
#pragma unroll
  for (int w = 0; w < 8; ++w) { ts += red[0][w]; ts2 += red[1][w]; }
  const float mu  = ts * (1.0f / G_);
  const float var = ts2 * (1.0f / G_) - mu * mu;
  const float rs  = rsqrtf(var + 1e-5f);
  float* dst = xn + (size_t)b * G_ + t * 16;
#pragma unroll
  for (int j = 0; j < 4; ++j) {
    float4 o;
    o.x = (v[j].x - mu) * rs; o.y = (v[j].y - mu) * rs;
    o.z = (v[j].z - mu) * rs; o.w = (v[j].w - mu) * rs;
    *(float4*)(dst + j * 4) = o;
  }
}

// ---------------------------------------------------------------------------
// Phase 2: per-head GEMM1 + ln affine fold + SiLU.
// grid = 32 heads * 8 d-blocks (64 cols each); block = 256 threads (8 waves).
// Each WG keeps all 256 batch rows -> W1 streamed from HBM exactly once.
// ---------------------------------------------------------------------------
__global__ __launch_bounds__(256) void gemm1_kernel(const float* __restrict__ xn,
                                                    const float* __restrict__ lnw,
                                                    const float* __restrict__ lnb,
                                                    const float* __restrict__ W1,
                                                    const float* __restrict__ b1,
                                                    float* __restrict__ hbuf) {
  const int n    = blockIdx.x >> 3;
  const int d0   = (blockIdx.x & 7) * 64;
  const int t    = threadIdx.x;
  const int lane = t & 31;
  const int wv   = t >> 5;
  const int dg   = wv & 1;           // 2 d-groups of 2 tiles
  const int bg   = wv >> 1;          // 4 b-groups of 4 tiles
  const int m    = lane & 15;        // row-in-tile (A) / col-in-tile (B,D)
  const int kh   = (lane >> 4) << 1; // K pair offset per WMMA layout

  __shared__ float sA[256 * LDA];    // xn tile   [b][k]
  __shared__ float sB[64 * LDA];     // W1 tile^T [d][k]
  __shared__ float sLw[KB];
  __shared__ float sLb[KB];

  v8f acc[4][2];
#pragma unroll
  for (int i = 0; i < 4; ++i) {
    acc[i][0] = v8f{0, 0, 0, 0, 0, 0, 0, 0};
    acc[i][1] = v8f{0, 0, 0, 0, 0, 0, 0, 0};
  }
  float c1x[2] = {0.f, 0.f}, c1y[2] = {0.f, 0.f};

  const size_t w1base = (size_t)n * G_ * H2_;

  for (int g0 = 0; g0 < G_; g0 += KB) {
    __syncthreads();
    {
      const int q  = t & 7;
      const int gl = t >> 3;
      // stage xn[0..255][g0..g0+31]  (coalesced 128B per 8 threads)
#pragma unroll
      for (int p = 0; p < 8; ++p) {
        const int row = gl + p * 32;
        const float4 v = *(const float4*)(xn + (size_t)row * G_ + g0 + q * 4);
        float* dst = sA + row * LDA + q * 4;
        dst[0] = v.x; dst[1] = v.y; dst[2] = v.z; dst[3] = v.w;
      }
      // stage W1 (transposed into [d][k])
#pragma unroll
      for (int p = 0; p < 2; ++p) {
        const int qq = q + p * 8;
        const float4 v =
            *(const float4*)(W1 + w1base + (size_t)(g0 + gl) * H2_ + d0 + qq * 4);
        sB[(qq * 4 + 0) * LDA + gl] = v.x;
        sB[(qq * 4 + 1) * LDA + gl] = v.y;
        sB[(qq * 4 + 2) * LDA + gl] = v.z;
        sB[(qq * 4 + 3) * LDA + gl] = v.w;
      }
      if (t < KB) {
        sLw[t] = lnw[(size_t)n * G_ + g0 + t];
        sLb[t] = lnb[(size_t)n * G_ + g0 + t];
      }
      if (g0 + KB < G_)  // hint next W1 K-block toward L2 (global_prefetch_b8)
        __builtin_prefetch(W1 + w1base + (size_t)(g0 + KB + gl) * H2_ + d0 + q * 8, 0, 0);
    }
    __syncthreads();

#pragma unroll
    for (int kc = 0; kc < KB / 4; ++kc) {
      const int kk = kc * 4 + kh;
      const float lwx = sLw[kk], lwy = sLw[kk + 1];
      const float lbx = sLb[kk], lby = sLb[kk + 1];
      v2f bf[2];
#pragma unroll
      for (int dt = 0; dt < 2; ++dt) {
        const int dl = dg * 32 + dt * 16 + m;
        bf[dt] = *(const v2f*)(sB + dl * LDA + kk);
        c1x[dt] += lbx * bf[dt].x;   // ln_b @ W1, in-lane partial
        c1y[dt] += lby * bf[dt].y;
      }
#pragma unroll
      for (int bt = 0; bt < 4; ++bt) {
        const int bl = bg * 64 + bt * 16 + m;
        v2f af = *(const v2f*)(sA + bl * LDA + kk);
        af.x *= lwx;                  // fold ln_w into A fragment
        af.y *= lwy;
#pragma unroll
        for (int dt = 0; dt < 2; ++dt)
          acc[bt][dt] = __builtin_amdgcn_wmma_f32_16x16x4_f32(
              false, af, false, bf[dt], (short)0, acc[bt][dt], false, false);
      }
    }
  }

  // epilogue: finish ln_b@W1 reduction, + b1, SiLU, store h[n][b][d]
#pragma unroll
  for (int dt = 0; dt < 2; ++dt) {
    float c = c1x[dt] + c1y[dt];
    c += __shfl_xor(c, 16, 32);       // combine K halves (lanes L and L^16 share d)
    const int d = d0 + dg * 32 + dt * 16 + m;
    const float bias = c + b1[n * H2_ + d];
#pragma unroll
    for (int bt = 0; bt < 4; ++bt) {
#pragma unroll
      for (int r = 0; r < 8; ++r) {
        const float x  = acc[bt][dt][r] + bias;
        const float hs = x / (1.0f + __expf(-x));    // SiLU
        const int b = bg * 64 + bt * 16 + r + ((lane >> 4) << 3);
        hbuf[((size_t)n * B_ + b) * H2_ + d] = hs;
      }
    }
  }
}

// ---------------------------------------------------------------------------
// Phase 3: per-head GEMM2 + fused bias/query_emb/seq_pool adds.
// grid = 32 heads * 4 h-blocks (64 cols each).
// ---------------------------------------------------------------------------
__global__ __launch_bounds__(256) void gemm2_kernel(const float* __restrict__ hbuf,
                                                    const float* __restrict__ W2,
                                                    const float* __restrict__ b2,
                                                    const float* __restrict__ qe,
                                                    const float* __restrict__ sp,
                                                    float* __restrict__ out) {
  const int n    = blockIdx.x >> 2;
  const int h0   = (blockIdx.x & 3) * 64;
  const int t    = threadIdx.x;
  const int lane = t & 31;
  const int wv   = t >> 5;
  const int hg   = wv & 1;
  const int bg   = wv >> 1;
  const int m    = lane & 15;
  const int kh   = (lane >> 4) << 1;

  __shared__ float sA[256 * LDA];    // h tile    [b][k]
  __shared__ float sB[64 * LDA];     // W2 tile^T [h][k]

  v8f acc[4][2];
#pragma unroll
  for (int i = 0; i < 4; ++i) {
    acc[i][0] = v8f{0, 0, 0, 0, 0, 0, 0, 0};
    acc[i][1] = v8f{0, 0, 0, 0, 0, 0, 0, 0};
  }

  const size_t w2base = (size_t)n * H2_ * H_;
  const size_t hbase  = (size_t)n * B_ * H2_;

  for (int g0 = 0; g0 < H2_; g0 += KB) {
    __syncthreads();
    {
      const int q  = t & 7;
      const int gl = t >> 3;
#pragma unroll
      for (int p = 0; p < 8; ++p) {
        const int row = gl + p * 32;
        const float4 v = *(const float4*)(hbuf + hbase + (size_t)row * H2_ + g0 + q * 4);
        float* dst = sA + row * LDA + q * 4;
        dst[0] = v.x; dst[1] = v.y; dst[2] = v.z; dst[3] = v.w;
      }
#pragma unroll
      for (int p = 0; p < 2; ++p) {
        const int qq = q + p * 8;
        const float4 v =
            *(const float4*)(W2 + w2base + (size_t)(g0 + gl) * H_ + h0 + qq * 4);
        sB[(qq * 4 + 0) * LDA + gl] = v.x;
        sB[(qq * 4 + 1) * LDA + gl] = v.y;
        sB[(qq * 4 + 2) * LDA + gl] = v.z;
        sB[(qq * 4 + 3) * LDA + gl] = v.w;
      }
    }
    __syncthreads();

#pragma unroll
    for (int kc = 0; kc < KB / 4; ++kc) {
      const int kk = kc * 4 + kh;
      v2f bf[2];
#pragma unroll
      for (int ht = 0; ht < 2; ++ht) {
        const int hl = hg * 32 + ht * 16 + m;
        bf[ht] = *(const v2f*)(sB + hl * LDA + kk);
      }
#pragma unroll
      for (int bt = 0; bt < 4; ++bt) {
        const int bl = bg * 64 + bt * 16 + m;
        const v2f af = *(const v2f*)(sA + bl * LDA + kk);
#pragma unroll
        for (int ht = 0; ht < 2; ++ht)
          acc[bt][ht] = __builtin_amdgcn_wmma_f32_16x16x4_f32(
              false, af, false, bf[ht], (short)0, acc[bt][ht], false, false);
      }
    }
  }

  const int s = n >> 2;
#pragma unroll
  for (int ht = 0; ht < 2; ++ht) {
    const int hc = h0 + hg * 32 + ht * 16 + m;
    const float badd = b2[n * H_ + hc] + qe[n * H_ + hc];
#pragma unroll
    for (int bt = 0; bt < 4; ++bt) {
#pragma unroll
      for (int r = 0; r < 8; ++r) {
        const int b = bg * 64 + bt * 16 + r + ((lane >> 4) << 3);
        const float v = acc[bt][ht][r] + badd + sp[((size_t)b * S_ + s) * H_ + hc];
        out[((size_t)b * N_ + n) * H_ + hc] = v;  // [B,S,Q,H] flat, n = s*Q+q
      }
    }
  }
}

extern "C" void kernel_launch(void* const* d_in, const int* in_sizes, int n_in,
                              void* d_out, int out_size, void* d_ws, size_t ws_size,
                              hipStream_t stream) {
  (void)in_sizes; (void)n_in; (void)out_size; (void)ws_size;
  const float* ns  = (const float*)d_in[0];
  const float* sp  = (const float*)d_in[1];
  const float* lnw = (const float*)d_in[2];
  const float* lnb = (const float*)d_in[3];
  const float* W1  = (const float*)d_in[4];
  const float* b1  = (const float*)d_in[5];
  const float* W2  = (const float*)d_in[6];
  const float* b2  = (const float*)d_in[7];
  const float* qe  = (const float*)d_in[8];
  float* out  = (float*)d_out;
  float* xn   = (float*)d_ws;               // 256*4096 f32 = 4 MB
  float* hbuf = xn + (size_t)B_ * G_;       // 32*256*512 f32 = 16 MB

  ln_kernel<<<B_, 256, 0, stream>>>(ns, sp, xn);
  gemm1_kernel<<<N_ * 8, 256, 0, stream>>>(xn, lnw, lnb, W1, b1, hbuf);
  gemm2_kernel<<<N_ * 4, 256, 0, stream>>>(hbuf, W2, b2, qe, sp, out);
}